// GCNLayer_56453050138798
// MI455X (gfx1250) — compile-verified
//
#include <hip/hip_runtime.h>

typedef __attribute__((ext_vector_type(2))) float v2f;
typedef __attribute__((ext_vector_type(4))) float v4f;
typedef __attribute__((ext_vector_type(8))) float v8f;

#define D_FEAT 64

__device__ __forceinline__ int bperm_i(int src, int lane) {
    return __builtin_amdgcn_ds_bpermute(lane << 2, src);
}
__device__ __forceinline__ float bperm_f(float src, int lane) {
    return __int_as_float(__builtin_amdgcn_ds_bpermute(lane << 2, __float_as_int(src)));
}

// ---------------------------------------------------------------------------
// Kernel 1: zero-initialize the output accumulator (harness poisons d_out).
// ---------------------------------------------------------------------------
__global__ void gcn_zero_kernel(float* __restrict__ out, int n) {
    int i = (blockIdx.x * blockDim.x + threadIdx.x) * 4;
    if (i + 3 < n) {
        *(v4f*)(out + i) = (v4f){0.f, 0.f, 0.f, 0.f};
    } else {
        for (int k = i; k < n; ++k) out[k] = 0.f;
    }
}

// ---------------------------------------------------------------------------
// Kernel 2: WMMA tile-SpMM scatter.
// One wave32 per tile of 16 edges. For the tile:
//   A[i][j] = val_j * (row_j == row_i)       (16x16, built as 4 chunks 16x4)
//   G[j][f] = embeds[col_j][f]               (16x64, as 4 column tiles 16x16)
//   P = A x G  -> P[i][:] = full segment sum for row_i within the tile
// Scatter P row i to out[row_i] only for the first occurrence of row_i.
// The scatter is branchless: non-scattering lanes atomically add +0.0f to
// a clamped (row 0) address, which is a numerical no-op, so no exec-mask
// save/restore sequences are needed around the 32 atomics.
// ---------------------------------------------------------------------------
__global__ __launch_bounds__(256) void gcn_spmm_wmma_kernel(
    const int* __restrict__ rows, const int* __restrict__ cols,
    const float* __restrict__ vals, const float* __restrict__ embeds,
    float* __restrict__ out, int nedges) {

    const int lane = threadIdx.x & 31;
    const int tile = blockIdx.x * 8 + (threadIdx.x >> 5);
    const int base = tile * 16;
    if (base >= nedges) return;          // wave-uniform: EXEC stays all-ones

    const int j = lane & 15;             // edge slot within tile (both halves)
    const int e = base + j;
    const bool valid = (e < nedges);
    int   row = valid ? rows[e] : (-1 - j);  // unique negative => never matches
    int   col = valid ? cols[e] : 0;
    float val = valid ? vals[e] : 0.0f;      // zero contribution when padded

    // First-occurrence flag for this tile slot's row value.
    int fo = 1;
#pragma unroll
    for (int i = 0; i < 15; ++i) {
        int ri = bperm_i(row, i);
        if (i < j && ri == row) fo = 0;
    }

    // Build A chunks. f32 16x4 A layout: lanes 0-15 hold M=0..15 with
    // VGPR0=K0, VGPR1=K1; lanes 16-31 hold K2,K3. khalf selects the K pair.
    const int khalf = (lane >> 4) << 1;      // 0 for lanes 0-15, 2 for 16-31
    v2f A[4];
    int ck0[4], ck1[4];
#pragma unroll
    for (int q = 0; q < 4; ++q) {
        const int k0 = 4 * q + khalf;
        const int k1 = k0 + 1;
        int   rk0 = bperm_i(row, k0);
        int   rk1 = bperm_i(row, k1);
        float vk0 = bperm_f(val, k0);
        float vk1 = bperm_f(val, k1);
        A[q].x = (rk0 == row) ? vk0 : 0.0f;
        A[q].y = (rk1 == row) ? vk1 : 0.0f;
        ck0[q] = bperm_i(col, k0);
        ck1[q] = bperm_i(col, k1);
    }

    // Scatter metadata for C/D rows. 16x16 f32 C/D layout: VGPR r holds
    // M=r (lanes 0-15) and M=r+8 (lanes 16-31), N = lane&15.
    int rowM[8], foM[8];
#pragma unroll
    for (int r = 0; r < 8; ++r) {
        const int M = (lane < 16) ? r : (r + 8);
        rowM[r] = bperm_i(row, M);
        foM[r]  = bperm_i(fo, M);
    }

    // Compute all four 16-feature column tiles.  Each needs K=16 => 4 chained
    // K=4 WMMAs.  B (4x16 f32) layout: lane = N, VGPR0 = K0 (lanes 0-15) /
    // K2 (lanes 16-31), VGPR1 = K1 / K3 -- same striping as A and C/D.
    v8f acc4[4];
#pragma unroll
    for (int c = 0; c < 4; ++c) {
        v8f acc = {0.f, 0.f, 0.f, 0.f, 0.f, 0.f, 0.f, 0.f};
#pragma unroll
        for (int q = 0; q < 4; ++q) {
            v2f B;
            B.x = embeds[ck0[q] * D_FEAT + c * 16 + j];
            B.y = embeds[ck1[q] * D_FEAT + c * 16 + j];
            acc = __builtin_amdgcn_wmma_f32_16x16x4_f32(
                /*neg_a=*/false, A[q], /*neg_b=*/false, B,
                /*c_mod=*/(short)0, acc, /*reuse_a=*/false, /*reuse_b=*/false);
        }
        acc4[c] = acc;
    }

    // Branchless scatter: one 64-bit base address per output row, constant
    // byte offsets (folded into the atomic's IOFFSET) for the 4 column tiles.
    // Non-scattering lanes add +0.0f at a clamped address (numerical no-op).
#pragma unroll
    for (int r = 0; r < 8; ++r) {
        const bool sc = (foM[r] != 0) && (rowM[r] >= 0);
        float* p = out + (sc ? rowM[r] : 0) * D_FEAT + j;
#pragma unroll
        for (int c = 0; c < 4; ++c) {
            atomicAdd(p + c * 16, sc ? acc4[c][r] : 0.0f);
        }
    }
}

// ---------------------------------------------------------------------------
// Launch: inputs are (rows, cols, edge_vals, embeds); output [N_NODES, 64] f32.
// ---------------------------------------------------------------------------
extern "C" void kernel_launch(void* const* d_in, const int* in_sizes, int n_in,
                              void* d_out, int out_size, void* d_ws, size_t ws_size,
                              hipStream_t stream) {
    const int*   rows   = (const int*)d_in[0];
    const int*   cols   = (const int*)d_in[1];
    const float* vals   = (const float*)d_in[2];
    const float* embeds = (const float*)d_in[3];
    float*       out    = (float*)d_out;

    const int nedges = in_sizes[0];

    // Zero the accumulator.
    {
        int nthreads = (out_size + 3) / 4;
        int blocks   = (nthreads + 255) / 256;
        gcn_zero_kernel<<<blocks, 256, 0, stream>>>(out, out_size);
    }

    // Tile scatter: 16 edges per wave, 8 waves (256 threads) per block.
    {
        int tiles  = (nedges + 15) / 16;
        int blocks = (tiles + 7) / 8;
        gcn_spmm_wmma_kernel<<<blocks, 256, 0, stream>>>(
            rows, cols, vals, embeds, out, nedges);
    }
}